// NetworkFeatureExtractor_858993459726
// MI455X (gfx1250) — compile-verified
//
#include <hip/hip_runtime.h>
#include <hip/hip_bf16.h>

// ---------------------------------------------------------------------------
// GNN feature extractor for MI455X (gfx1250, wave32).
// GEMMs use V_WMMA_F32_16X16X4_F32 (fp32 matrix cores, matches fp32 reference).
// Edge aggregation uses global_atomic_add_f32 (L2-resident: whole working set
// ~90MB fits in the 192MB L2).
// ---------------------------------------------------------------------------

#define NN 50000
#define NE 800000

typedef float v2f __attribute__((ext_vector_type(2)));
typedef float v4f __attribute__((ext_vector_type(4)));
typedef float v8f __attribute__((ext_vector_type(8)));

// ---- degree count: deg[col[e]] += 1 ---------------------------------------
__global__ __launch_bounds__(256)
void deg_kernel(const long long* __restrict__ col, float* __restrict__ deg, int E)
{
    int e = blockIdx.x * 256 + threadIdx.x;
    if (e < E) atomicAdd(&deg[(int)col[e]], 1.0f);
}

// ---- dinv = deg>0 ? rsqrt(deg) : 0 ----------------------------------------
__global__ __launch_bounds__(256)
void dinv_kernel(const float* __restrict__ deg, float* __restrict__ dinv, int n)
{
    int i = blockIdx.x * 256 + threadIdx.x;
    if (i < n) {
        float d = deg[i];
        dinv[i] = (d > 0.0f) ? rsqrtf(d) : 0.0f;
    }
}

// ---------------------------------------------------------------------------
// Dual GEMM via WMMA f32 16x16x4.  For a 16-row tile of A [N,128]:
//   dst1[r, col] = relu(A@W1 + b1)   (row stride sd1; FC branch)
//   dst2[r, col] = A@W2              (row stride 64;  GCN pre-aggregation h)
// Block = 128 threads = 4 waves; wave w owns output columns [16w, 16w+16).
// A tile staged in LDS (row stride 132 dwords: 132%64=4 -> conflict-free
// 16-lane reads; 16B-aligned float4 stores).
//
// Fragment layouts per CDNA5 ISA 7.12.2 (wave32):
//   A (16x4 f32, 2 VGPRs): lane half h=lane>>4, m=lane&15, reg j -> A[m][2h+j]
//   B (4x16 f32, 2 VGPRs): reg r, half h -> B[2h+r][lane&15]
//   C/D (16x16 f32, 8 VGPRs): reg r, half h -> D[r+8h][lane&15]
// ---------------------------------------------------------------------------
__global__ __launch_bounds__(128)
void dual_gemm_kernel(const float* __restrict__ A,
                      const float* __restrict__ W1, const float* __restrict__ b1,
                      const float* __restrict__ W2,
                      float* __restrict__ dst1, int sd1,
                      float* __restrict__ dst2)
{
    __shared__ float As[16][132];

    const int tid = threadIdx.x;
    const int rb  = blockIdx.x * 16;        // base node row of this tile

    // cooperative, fully-coalesced load of the 16x128 A tile
#pragma unroll
    for (int i = 0; i < 4; ++i) {
        int f = i * 512 + tid * 4;          // flat float index into 16x128 tile
        int r = f >> 7;
        int c = f & 127;
        v4f v = *reinterpret_cast<const v4f*>(A + (long)(rb + r) * 128 + c);
        *reinterpret_cast<v4f*>(&As[r][c]) = v;
    }
    __syncthreads();

    const int wave = tid >> 5;
    const int lane = tid & 31;
    const int h    = lane >> 4;             // lane half
    const int m    = lane & 15;            // A row within tile / B,D column idx
    const int colb = wave * 16 + m;        // output column owned by this lane

    v8f acc1 = {};
    v8f acc2 = {};

#pragma unroll
    for (int k = 0; k < 128; k += 4) {
        const int ka = k + 2 * h;
        v2f a = *reinterpret_cast<const v2f*>(&As[m][ka]);

        v2f bf1, bf2;
        bf1.x = W1[(long)ka * 64 + colb];
        bf1.y = W1[(long)(ka + 1) * 64 + colb];
        bf2.x = W2[(long)ka * 64 + colb];
        bf2.y = W2[(long)(ka + 1) * 64 + colb];

        acc1 = __builtin_amdgcn_wmma_f32_16x16x4_f32(false, a, false, bf1,
                                                     (short)0, acc1, false, false);
        acc2 = __builtin_amdgcn_wmma_f32_16x16x4_f32(false, a, false, bf2,
                                                     (short)0, acc2, false, false);
    }

    const float bias = b1[colb];
#pragma unroll
    for (int r = 0; r < 8; ++r) {
        const int mr = r + 8 * h;
        float v1 = acc1[r] + bias;
        dst1[(long)(rb + mr) * sd1 + colb] = (v1 > 0.0f) ? v1 : 0.0f;
        dst2[(long)(rb + mr) * 64 + colb]  = acc2[r];
    }
}

// ---- edge scatter: agg[col[e], f] += h[row[e], f] * dinv[row]*dinv[col] ----
__global__ __launch_bounds__(256)
void gcn_scatter_kernel(const float* __restrict__ h,
                        const long long* __restrict__ row,
                        const long long* __restrict__ col,
                        const float* __restrict__ dinv,
                        float* __restrict__ agg, int E)
{
    int idx = blockIdx.x * 256 + threadIdx.x;   // 4 edges x 64 feats per block
    int e = idx >> 6;
    int f = idx & 63;
    if (e >= E) return;
    int r = (int)row[e];
    int c = (int)col[e];
    float nrm = dinv[r] * dinv[c];
    atomicAdd(&agg[(long)c * 64 + f], h[(long)r * 64 + f] * nrm);
}

// ---- cur[:, 64:128] = relu(agg + b) ---------------------------------------
__global__ __launch_bounds__(256)
void bias_relu_kernel(const float* __restrict__ agg, const float* __restrict__ b,
                      float* __restrict__ dst, int n)
{
    int idx = blockIdx.x * 256 + threadIdx.x;
    if (idx >= n * 64) return;
    int i = idx >> 6;
    int f = idx & 63;
    float v = agg[idx] + b[f];
    dst[(long)i * 128 + 64 + f] = (v > 0.0f) ? v : 0.0f;
}

// ---- out += relu(agg + b) -------------------------------------------------
__global__ __launch_bounds__(256)
void final_add_kernel(const float* __restrict__ agg, const float* __restrict__ b,
                      float* __restrict__ out, int n)
{
    int idx = blockIdx.x * 256 + threadIdx.x;
    if (idx >= n * 64) return;
    int f = idx & 63;
    float v = agg[idx] + b[f];
    out[idx] += (v > 0.0f) ? v : 0.0f;
}

// ---------------------------------------------------------------------------
extern "C" void kernel_launch(void* const* d_in, const int* in_sizes, int n_in,
                              void* d_out, int out_size, void* d_ws, size_t ws_size,
                              hipStream_t stream)
{
    const float*      x     = (const float*)d_in[0];
    const long long*  ei    = (const long long*)d_in[1];   // int64 [2, E]
    const float*      W_fc1 = (const float*)d_in[2];
    const float*      b_fc1 = (const float*)d_in[3];
    const float*      W_gc1 = (const float*)d_in[4];
    const float*      b_gc1 = (const float*)d_in[5];
    const float*      W_fcm = (const float*)d_in[6];
    const float*      b_fcm = (const float*)d_in[7];
    const float*      W_gcm = (const float*)d_in[8];
    const float*      b_gcm = (const float*)d_in[9];
    const float*      W_fcf = (const float*)d_in[10];
    const float*      b_fcf = (const float*)d_in[11];
    const float*      W_gcf = (const float*)d_in[12];
    const float*      b_gcf = (const float*)d_in[13];
    float*            out   = (float*)d_out;

    const long long* row = ei;
    const long long* col = ei + NE;

    // scratch carve-out (~77 MB, 256B aligned slices)
    char* ws = (char*)d_ws;
    auto alloc = [&](size_t bytes) -> char* {
        char* p = ws;
        ws += (bytes + 255) & ~(size_t)255;
        return p;
    };
    float* deg  = (float*)alloc((size_t)NN * 4);
    float* dinv = (float*)alloc((size_t)NN * 4);
    float* cur  = (float*)alloc((size_t)NN * 128 * 4);
    float* cur2 = (float*)alloc((size_t)NN * 128 * 4);
    float* hbuf = (float*)alloc((size_t)NN * 64 * 4);
    float* agg  = (float*)alloc((size_t)NN * 64 * 4);

    const int nTileBlocks = NN / 16;                   // 3125
    const int nEdgeBlocks = (NE + 255) / 256;          // 3125
    const int nEFBlocks   = (int)(((long)NE * 64 + 255) / 256);  // 200000
    const int nNFBlocks   = (NN * 64 + 255) / 256;     // 12500

    // ---- normalization ----
    hipMemsetAsync(deg, 0, (size_t)NN * 4, stream);
    deg_kernel<<<nEdgeBlocks, 256, 0, stream>>>(col, deg, NE);
    dinv_kernel<<<(NN + 255) / 256, 256, 0, stream>>>(deg, dinv, NN);

    // ---- layer 1: x[N,128] -> cur[N,128] ----
    dual_gemm_kernel<<<nTileBlocks, 128, 0, stream>>>(x, W_fc1, b_fc1, W_gc1,
                                                      cur, 128, hbuf);
    hipMemsetAsync(agg, 0, (size_t)NN * 64 * 4, stream);
    gcn_scatter_kernel<<<nEFBlocks, 256, 0, stream>>>(hbuf, row, col, dinv, agg, NE);
    bias_relu_kernel<<<nNFBlocks, 256, 0, stream>>>(agg, b_gc1, cur, NN);

    // ---- mid layer: cur[N,128] -> cur2[N,128] ----
    dual_gemm_kernel<<<nTileBlocks, 128, 0, stream>>>(cur, W_fcm, b_fcm, W_gcm,
                                                      cur2, 128, hbuf);
    hipMemsetAsync(agg, 0, (size_t)NN * 64 * 4, stream);
    gcn_scatter_kernel<<<nEFBlocks, 256, 0, stream>>>(hbuf, row, col, dinv, agg, NE);
    bias_relu_kernel<<<nNFBlocks, 256, 0, stream>>>(agg, b_gcm, cur2, NN);

    // ---- final layer: cur2[N,128] -> out[N,64] ----
    dual_gemm_kernel<<<nTileBlocks, 128, 0, stream>>>(cur2, W_fcf, b_fcf, W_gcf,
                                                      out, 64, hbuf);
    hipMemsetAsync(agg, 0, (size_t)NN * 64 * 4, stream);
    gcn_scatter_kernel<<<nEFBlocks, 256, 0, stream>>>(hbuf, row, col, dinv, agg, NE);
    final_add_kernel<<<nNFBlocks, 256, 0, stream>>>(agg, b_gcf, out, NN);
}